// QLoRAEmbedding_10505490006719
// MI455X (gfx1250) — compile-verified
//
#include <hip/hip_runtime.h>

// QLoRA embedding: out[t, e] = base_weight[e, x[t]] + scaling * (lora_A[x[t]] @ lora_B)[e]
//
// Memory-bound kernel: 64 MiB NT output stream + 128 MiB L2-resident transposed
// gather of base_weight. LoRA GEMM (rank 16) runs on the WMMA pipe with exact
// fp32 math (V_WMMA_F32_16X16X4_F32), overlapped with the scattered gathers.

typedef __attribute__((ext_vector_type(2))) float v2f;
typedef __attribute__((ext_vector_type(8))) float v8f;

#define LORA_RANK 16

__global__ __launch_bounds__(256) void qlora_embed_kernel(
    const int*   __restrict__ x,       // [n_tokens]
    const float* __restrict__ base_w,  // [embed, vocab]  (row-major)
    const float* __restrict__ lora_A,  // [vocab, RANK]
    const float* __restrict__ lora_B,  // [RANK, embed]
    float*       __restrict__ out,     // [n_tokens, embed]
    int embed, int vocab, float scaling)
{
    __shared__ int s_tok[16];

    const int tid  = threadIdx.x;
    const int lane = tid & 31;
    const int wave = tid >> 5;              // 0..7
    const int m0   = blockIdx.x * 16;       // token-tile base

    if (tid < 16) s_tok[tid] = x[m0 + tid];
    __syncthreads();

    const int half = lane >> 4;             // 0: lanes 0-15, 1: lanes 16-31
    const int l16  = lane & 15;

    // ---- A fragments for V_WMMA_F32_16X16X4_F32 (built once per wave) ----
    // 32-bit A 16x4 layout: lanes 0-15 hold {K=kc, kc+1}, lanes 16-31 {K=kc+2, kc+3},
    // M = lane % 16.  Rows of lora_A are contiguous -> float2 loads.
    const long tokM = s_tok[l16];
    v2f a[4];
#pragma unroll
    for (int kk = 0; kk < 4; ++kk) {
        const int k0 = kk * 4 + half * 2;
        const float* ap = lora_A + tokM * LORA_RANK + k0;
        a[kk].x = ap[0] * scaling;          // fold alpha/rank into A
        a[kk].y = ap[1] * scaling;
    }

    const int tilesN       = embed >> 4;    // 16-wide N tiles
    const int tilesPerWave = tilesN >> 3;   // 8 waves per block

    for (int t = 0; t < tilesPerWave; ++t) {
        const int n0 = (wave * tilesPerWave + t) * 16;
        const int e  = n0 + l16;            // this lane's embed column

        // ---- scattered base gather (issued first; overlaps the WMMAs) ----
        // g[r] pairs with D-fragment element r: row M = r + 8*half, col N = l16.
        float g[8];
#pragma unroll
        for (int r = 0; r < 8; ++r) {
            const long tok = s_tok[r + half * 8];
            g[r] = base_w[(long)e * vocab + tok];   // L2-resident, each lane its own line
        }

        // ---- LoRA tile: D = A(16x16) x B(16x16) via 4 x 16x16x4 f32 WMMA ----
        // B 4x16 layout mirrors A: lanes 0-15 hold {K=kc,kc+1}, lanes 16-31
        // {K=kc+2,kc+3}, N = lane % 16. lora_B row-major -> two coalesced 64B
        // half-wave reads per fragment register.
        v8f c = {};
#pragma unroll
        for (int kk = 0; kk < 4; ++kk) {
            const int k0 = kk * 4 + half * 2;
            v2f b;
            b.x = lora_B[(long)k0       * embed + e];
            b.y = lora_B[(long)(k0 + 1) * embed + e];
            // 8 args: (neg_a, A, neg_b, B, c_mod, C, reuse_a, reuse_b)
            c = __builtin_amdgcn_wmma_f32_16x16x4_f32(false, a[kk], false, b,
                                                      (short)0, c, false, false);
        }

        // ---- fuse base + lora, stream out with NT stores (write-once data) ----
        // D layout: VGPR r -> M = r + 8*half, N = l16. Per r each half-wave
        // writes 16 consecutive floats (64B segments).
#pragma unroll
        for (int r = 0; r < 8; ++r) {
            const int m = r + half * 8;
            __builtin_nontemporal_store(g[r] + c[r],
                                        out + (long)(m0 + m) * embed + n0 + l16);
        }
    }
}

extern "C" void kernel_launch(void* const* d_in, const int* in_sizes, int n_in,
                              void* d_out, int out_size, void* d_ws, size_t ws_size,
                              hipStream_t stream) {
    const int*   x      = (const int*)d_in[0];     // [4, 4096] int32
    const float* base_w = (const float*)d_in[1];   // [EMBED, VOCAB] fp32
    const float* lora_A = (const float*)d_in[2];   // [VOCAB, RANK] fp32
    const float* lora_B = (const float*)d_in[3];   // [RANK, EMBED] fp32
    float*       out    = (float*)d_out;           // [4, 4096, EMBED] fp32

    const int n_tokens = in_sizes[0];              // 16384
    const int embed    = out_size / n_tokens;      // 1024
    const int vocab    = in_sizes[1] / embed;      // 32000
    const float scaling = 16.0f / (float)LORA_RANK; // ALPHA / RANK = 1.0

    const int blocks = n_tokens / 16;              // 1024 token tiles
    qlora_embed_kernel<<<blocks, 256, 0, stream>>>(x, base_w, lora_A, lora_B,
                                                   out, embed, vocab, scaling);
}